// CrossAttention_90512140796621
// MI455X (gfx1250) — compile-verified
//
#include <hip/hip_runtime.h>
#include <hip/hip_bf16.h>
#include <stdint.h>

typedef __attribute__((ext_vector_type(2))) float v2f;
typedef __attribute__((ext_vector_type(8))) float v8f;

// Problem constants (from reference)
#define BATCH 2
#define CIN   192          // channels of x
#define INNER 512          // heads * dim_head
#define HEADS 8
#define DH    64
#define SDIM  32768        // 32*32*32 spatial
#define ATTN_SCALE 0.125f  // 64^-0.5

// ---------------------------------------------------------------------------
// CDNA5 async global->LDS copy (ASYNCcnt-tracked, no VGPR staging).
// GV mode: vdst = LDS byte offset VGPR, vaddr = 64-bit global address, no SADDR.
// ---------------------------------------------------------------------------
__device__ __forceinline__ void async_copy_b128(void* lds_dst, const void* gsrc) {
    uint32_t lds_off = (uint32_t)(uintptr_t)lds_dst;       // low 32 bits = LDS byte addr
    uint64_t gaddr   = (uint64_t)(uintptr_t)gsrc;
    asm volatile("global_load_async_to_lds_b128 %0, %1, off"
                 :: "v"(lds_off), "v"(gaddr)
                 : "memory");
}
#define ASYNC_WAIT(N) asm volatile("s_wait_asynccnt %0" :: "n"(N) : "memory")

// ---------------------------------------------------------------------------
// Kernel 0: zero the Gram accumulator (it is reduced with atomics).
// ---------------------------------------------------------------------------
__global__ void ca_zero_kernel(float* __restrict__ p, int n) {
    int i = blockIdx.x * blockDim.x + threadIdx.x;
    if (i < n) p[i] = 0.0f;
}

// ---------------------------------------------------------------------------
// Kernel 1: Gram matrix  G[b] = X_b * X_b^T   (192x192, K = 32768 split-K)
// 512 threads = 16 waves. Each WG handles 512 spatial columns in 64-col tiles,
// double-buffered in LDS via global_load_async_to_lds_b128. Wave (wr,wc) in a
// 4x4 grid owns a 3x3 block of 16x16 tiles. v_wmma_f32_16x16x4_f32,
// fp32-exact. Partial sums reduced via global fp32 atomics.
// ---------------------------------------------------------------------------
#define GR_KT      64            // spatial columns per LDS tile
#define GR_SPW     512           // spatial columns per workgroup
#define XS_STRIDE  68            // LDS row stride (bank-conflict-free, 16B rows)

__global__ __launch_bounds__(512) void ca_gram_kernel(const float* __restrict__ x,
                                                      float* __restrict__ G) {
    __shared__ float xs[2][CIN * XS_STRIDE];  // 2 x 52 KB double buffer

    const int b = blockIdx.y;
    const float* xb = x + (size_t)b * CIN * SDIM;
    const int s_base = blockIdx.x * GR_SPW;

    const int lane = threadIdx.x & 31;
    const int wave = threadIdx.x >> 5;
    const int wr = wave >> 2, wc = wave & 3;
    const int l16 = lane & 15, half = lane >> 4;

    // Per-thread cooperative async staging: 3072 B128 segments / 512 thr = 6.
    auto issue_tile = [&](int s0, int buf) {
        #pragma unroll
        for (int i = 0; i < 6; ++i) {
            int idx = threadIdx.x + i * 512;
            int row = idx >> 4, c4 = idx & 15;
            async_copy_b128(&xs[buf][row * XS_STRIDE + c4 * 4],
                            xb + (size_t)row * SDIM + s0 + c4 * 4);
        }
    };

    issue_tile(s_base, 0);   // prologue: tile 0 in flight

    v8f acc[3][3] = {};
    const int nchunks = GR_SPW / GR_KT;
    for (int ch = 0; ch < nchunks; ++ch) {
        if (ch + 1 < nchunks) {
            issue_tile(s_base + (ch + 1) * GR_KT, (ch + 1) & 1);
            ASYNC_WAIT(6);   // drain tile ch (in-order), keep tile ch+1 in flight
        } else {
            ASYNC_WAIT(0);
        }
        __syncthreads();

        const float* xt = xs[ch & 1];
        #pragma unroll 4
        for (int k0 = 0; k0 < GR_KT; k0 += 4) {
            // A fragment: 16x4 rows of X; B fragment: X^T columns (same values)
            v2f af[3], bf[3];
            #pragma unroll
            for (int t = 0; t < 3; ++t) {
                int ra = 16 * (3 * wr + t) + l16;
                af[t].x = xt[ra * XS_STRIDE + k0 + 2 * half];
                af[t].y = xt[ra * XS_STRIDE + k0 + 2 * half + 1];
                int rb = 16 * (3 * wc + t) + l16;
                bf[t].x = xt[rb * XS_STRIDE + k0 + 2 * half];
                bf[t].y = xt[rb * XS_STRIDE + k0 + 2 * half + 1];
            }
            #pragma unroll
            for (int i = 0; i < 3; ++i)
                #pragma unroll
                for (int j = 0; j < 3; ++j)
                    acc[i][j] = __builtin_amdgcn_wmma_f32_16x16x4_f32(
                        false, af[i], false, bf[j], (short)0, acc[i][j], false, false);
        }
        __syncthreads();     // protect buffer (ch+2)&1 before it is re-filled
    }

    // split-K reduction into G via fp32 global atomics
    float* Gb = G + (size_t)b * CIN * CIN;
    #pragma unroll
    for (int i = 0; i < 3; ++i)
        #pragma unroll
        for (int j = 0; j < 3; ++j) {
            int rb = 16 * (3 * wr + i), cb = 16 * (3 * wc + j);
            #pragma unroll
            for (int v = 0; v < 8; ++v) {
                int row = rb + v + 8 * half;
                int col = cb + l16;
                atomicAdd(&Gb[row * CIN + col], acc[i][j][v]);
            }
        }
}

// ---------------------------------------------------------------------------
// Kernel 2: per (b, head): sim = scale * Wq_h G Wk_h^T ; softmax ; AV = attn Wv_h
// All matrices tiny (<=64x192). 256 threads, LDS staging.
// ---------------------------------------------------------------------------
__global__ __launch_bounds__(256) void ca_attn_kernel(const float* __restrict__ G,
                                                      const float* __restrict__ wq,
                                                      const float* __restrict__ wk,
                                                      const float* __restrict__ wv,
                                                      float* __restrict__ AV) {
    __shared__ float T1[DH * CIN];   // 64*192*4 = 48 KB
    __shared__ float sim[DH * DH];   // 16 KB

    const int b = blockIdx.y, n = blockIdx.x;
    const float* Gb = G + (size_t)b * CIN * CIN;

    // T1 = Wq_h * G  [64 x 192]
    for (int e = threadIdx.x; e < DH * CIN; e += 256) {
        int i = e / CIN, c2 = e % CIN;
        const float* wqr = wq + (size_t)(n * DH + i) * CIN;
        float a = 0.f;
        #pragma unroll 4
        for (int c1 = 0; c1 < CIN; ++c1) a += wqr[c1] * Gb[c1 * CIN + c2];
        T1[i * CIN + c2] = a;
    }
    __syncthreads();

    // sim = scale * T1 * Wk_h^T  [64 x 64]
    for (int e = threadIdx.x; e < DH * DH; e += 256) {
        int i = e >> 6, j = e & 63;
        const float* wkr = wk + (size_t)(n * DH + j) * CIN;
        const float* t1r = &T1[i * CIN];
        float a = 0.f;
        #pragma unroll 4
        for (int c = 0; c < CIN; ++c) a += t1r[c] * wkr[c];
        sim[e] = a * ATTN_SCALE;
    }
    __syncthreads();

    // row softmax (64 rows, one thread each)
    if (threadIdx.x < DH) {
        float* row = &sim[threadIdx.x * DH];
        float m = row[0];
        #pragma unroll 4
        for (int j = 1; j < DH; ++j) m = fmaxf(m, row[j]);
        float s = 0.f;
        #pragma unroll 4
        for (int j = 0; j < DH; ++j) { float e2 = __expf(row[j] - m); row[j] = e2; s += e2; }
        float inv = 1.f / s;
        #pragma unroll 4
        for (int j = 0; j < DH; ++j) row[j] *= inv;
    }
    __syncthreads();

    // AV_h = attn * Wv_h  [64 x 192]  -> AV_flat[b][n*64+i][c]
    float* AVb = AV + ((size_t)b * INNER + n * DH) * CIN;
    for (int e = threadIdx.x; e < DH * CIN; e += 256) {
        int i = e / CIN, c = e % CIN;
        const float* ar = &sim[i * DH];
        float a = 0.f;
        #pragma unroll 4
        for (int j = 0; j < DH; ++j) a += ar[j] * wv[(size_t)(n * DH + j) * CIN + c];
        AVb[i * CIN + c] = a;
    }
}

// ---------------------------------------------------------------------------
// Kernel 3: W_eff[b] = wo @ AV_flat[b]   (192x192, K = 512)
// One wave per 16x16 tile, fp32 WMMA over K.
// ---------------------------------------------------------------------------
__global__ __launch_bounds__(32) void ca_weff_kernel(const float* __restrict__ wo,
                                                     const float* __restrict__ AV,
                                                     float* __restrict__ Weff) {
    const int b = blockIdx.y;
    const int ti = blockIdx.x / 12, tj = blockIdx.x % 12;
    const int lane = threadIdx.x;
    const int l16 = lane & 15, half = lane >> 4;
    const float* AVb = AV + (size_t)b * INNER * CIN;

    v8f acc = {};
    #pragma unroll 4
    for (int k0 = 0; k0 < INNER; k0 += 4) {
        v2f a, bf;
        int arow = 16 * ti + l16;
        a.x = wo[(size_t)arow * INNER + k0 + 2 * half];
        a.y = wo[(size_t)arow * INNER + k0 + 2 * half + 1];
        int bcol = 16 * tj + l16;
        bf.x = AVb[(size_t)(k0 + 2 * half) * CIN + bcol];
        bf.y = AVb[(size_t)(k0 + 2 * half + 1) * CIN + bcol];
        acc = __builtin_amdgcn_wmma_f32_16x16x4_f32(false, a, false, bf,
                                                    (short)0, acc, false, false);
    }
    float* Wb = Weff + (size_t)b * CIN * CIN;
    #pragma unroll
    for (int v = 0; v < 8; ++v) {
        int row = 16 * ti + v + 8 * half;
        Wb[row * CIN + 16 * tj + l16] = acc[v];
    }
}

// ---------------------------------------------------------------------------
// Kernel 4: out[b] = W_eff[b] @ X_b + bo   (192 x 32768, K = 192)
// W_eff staged in 150 KB LDS via async copies (WGP has 320 KB); X streamed
// coalesced from global (L2-backed). 16 waves in a 4x4 grid, 3x2 tiles each.
// ---------------------------------------------------------------------------
#define OUT_NCOLS 128
#define WF_STRIDE 196   // LDS row stride for W_eff (conflict-free, 16B rows)

__global__ __launch_bounds__(512) void ca_out_kernel(const float* __restrict__ x,
                                                     const float* __restrict__ Weff,
                                                     const float* __restrict__ bo,
                                                     float* __restrict__ out) {
    __shared__ float wsm[CIN * WF_STRIDE];  // 192*196*4 = 150528 B

    const int b = blockIdx.y;
    const int s0 = blockIdx.x * OUT_NCOLS;
    const float* xb = x + (size_t)b * CIN * SDIM;
    const float* Wb = Weff + (size_t)b * CIN * CIN;

    // async stage of W_eff: 9216 B128 segments / 512 thr = 18 each
    #pragma unroll
    for (int i = 0; i < 18; ++i) {
        int idx = threadIdx.x + i * 512;
        int row = idx / 48, c4 = idx % 48;
        async_copy_b128(&wsm[row * WF_STRIDE + c4 * 4], Wb + row * CIN + c4 * 4);
    }
    // warm the x slab while the DMA is in flight
    __builtin_prefetch(xb + (size_t)(threadIdx.x >> 2) * SDIM + s0 + (threadIdx.x & 3) * 32, 0, 3);
    ASYNC_WAIT(0);
    __syncthreads();

    const int lane = threadIdx.x & 31;
    const int wave = threadIdx.x >> 5;
    const int wr = wave >> 2, wc = wave & 3;
    const int l16 = lane & 15, half = lane >> 4;

    v8f acc[3][2] = {};

    #pragma unroll 2
    for (int k0 = 0; k0 < CIN; k0 += 4) {
        v2f af[3];
        #pragma unroll
        for (int t = 0; t < 3; ++t) {
            int arow = 16 * (3 * wr + t) + l16;
            af[t].x = wsm[arow * WF_STRIDE + k0 + 2 * half];
            af[t].y = wsm[arow * WF_STRIDE + k0 + 2 * half + 1];
        }
        v2f bf[2];
        #pragma unroll
        for (int t = 0; t < 2; ++t) {
            int scol = s0 + 16 * (2 * wc + t) + l16;
            bf[t].x = xb[(size_t)(k0 + 2 * half) * SDIM + scol];
            bf[t].y = xb[(size_t)(k0 + 2 * half + 1) * SDIM + scol];
        }
        #pragma unroll
        for (int i = 0; i < 3; ++i)
            #pragma unroll
            for (int j = 0; j < 2; ++j)
                acc[i][j] = __builtin_amdgcn_wmma_f32_16x16x4_f32(
                    false, af[i], false, bf[j], (short)0, acc[i][j], false, false);
    }

    float* ob = out + (size_t)b * CIN * SDIM;
    #pragma unroll
    for (int i = 0; i < 3; ++i)
        #pragma unroll
        for (int j = 0; j < 2; ++j) {
            int rb = 16 * (3 * wr + i), cb = 16 * (2 * wc + j);
            #pragma unroll
            for (int v = 0; v < 8; ++v) {
                int row = rb + v + 8 * half;
                ob[(size_t)row * SDIM + s0 + cb + l16] = acc[i][j][v] + bo[row];
            }
        }
}

// ---------------------------------------------------------------------------
// Host-side launcher
// ---------------------------------------------------------------------------
extern "C" void kernel_launch(void* const* d_in, const int* in_sizes, int n_in,
                              void* d_out, int out_size, void* d_ws, size_t ws_size,
                              hipStream_t stream) {
    const float* x  = (const float*)d_in[0];
    const float* wq = (const float*)d_in[1];
    const float* wk = (const float*)d_in[2];
    const float* wv = (const float*)d_in[3];
    const float* wo = (const float*)d_in[4];
    const float* bo = (const float*)d_in[5];
    float* out = (float*)d_out;

    float* ws = (float*)d_ws;
    const int G_ELEMS  = BATCH * CIN * CIN;        // 73728
    const int AV_ELEMS = BATCH * INNER * CIN;      // 196608
    float* G    = ws;
    float* AV   = ws + G_ELEMS;
    float* Weff = ws + G_ELEMS + AV_ELEMS;

    // 0) zero Gram accumulator (atomic split-K target)
    ca_zero_kernel<<<(G_ELEMS + 255) / 256, 256, 0, stream>>>(G, G_ELEMS);

    // 1) G_b = X_b X_b^T  (split-K: 64 chunks of 512 columns per batch)
    ca_gram_kernel<<<dim3(SDIM / GR_SPW, BATCH), 512, 0, stream>>>(x, G);

    // 2) per-(b, head) attention fold -> AV_flat
    ca_attn_kernel<<<dim3(HEADS, BATCH), 256, 0, stream>>>(G, wq, wk, wv, AV);

    // 3) W_eff = wo @ AV_flat
    ca_weff_kernel<<<dim3(144, BATCH), 32, 0, stream>>>(wo, AV, Weff);

    // 4) out = W_eff @ X + bo
    ca_out_kernel<<<dim3(SDIM / OUT_NCOLS, BATCH), 512, 0, stream>>>(x, Weff, bo, out);
}